// BEV2RV_36996848287966
// MI455X (gfx1250) — compile-verified
//
#include <hip/hip_runtime.h>
#include <stdint.h>

// BEV (4,64,512,512) fp32 -> range-view (4,64,64,2048) scatter-max.
// k1: init out = -inf ; k2: TDM-staged atomic scatter-max ; k3: -inf -> 0.

#define B_ 4
#define C_ 64
#define HB 512
#define WB 512
#define HR 64
#define WR 2048
#define NRV (HR * WR)          // 131072 bins per (b,c)
#define TILE_X 256

typedef uint32_t u32;
typedef uint64_t u64;
typedef u32 v4u __attribute__((ext_vector_type(4)));
typedef int  v8i __attribute__((ext_vector_type(8)));
typedef int  v4i __attribute__((ext_vector_type(4)));

// Native CDNA5 no-return fp32 max atomic, device scope (output is L2-resident:
// 128MB output < 192MB L2, so these RMWs never round-trip to HBM).
__device__ __forceinline__ void atomic_max_f32_dev(float* p, float v) {
    asm volatile("global_atomic_max_num_f32 %0, %1, off scope:SCOPE_DEV"
                 :
                 : "v"((u64)(uintptr_t)p), "v"(v)
                 : "memory");
}

__global__ __launch_bounds__(256) void rv_init_kernel(float4* __restrict__ out, int n4) {
    int i = blockIdx.x * 256 + threadIdx.x;
    if (i < n4) {
        const float NI = -__builtin_inff();
        out[i] = make_float4(NI, NI, NI, NI);
    }
}

__global__ __launch_bounds__(TILE_X) void bev2rv_scatter_kernel(
    const float* __restrict__ bev_feat,   // (B,C,HB,WB)
    const int*   __restrict__ zbin,       // (B,1,HB,WB)
    float*       __restrict__ out)        // (B,C,HR,WR), pre-filled with -inf
{
    __shared__ float tile[C_ * TILE_X];   // 64 KB: 64 channel rows x 256 x-pos

    // Generic address of an LDS object: low 32 bits are the LDS byte offset
    // (aperture mapping). This is the lds_addr the TDM D# needs, and taking
    // it via ptrtoint also keeps/captures the LDS allocation.
    const u32 lds_base = (u32)(uintptr_t)tile;

    const int t   = threadIdx.x;
    const int blk = blockIdx.x;
    const int XT  = WB / TILE_X;          // 2 tiles per row
    const int xt  = blk % XT;
    const int y   = (blk / XT) % HB;
    const int b   = blk / (XT * HB);
    const int x0  = xt * TILE_X;

    // ---- Issue one TDM 2D tile load (wave 0 only; TDM ignores EXEC) ----
    // Tile: dim0 = 256 fp32 (contiguous x), dim1 = 64 (channels),
    // tensor_dim0_stride = HB*WB = 262144 elements between channel rows.
    if (t < 32) {
        u64 gaddr = (u64)(uintptr_t)(bev_feat
                      + (u64)b * C_ * HB * WB + (u64)y * WB + (u64)x0);
        v4u g0;
        g0.x = 1u;                                   // count=1, user descriptor
        g0.y = lds_base;                             // lds_addr = tile[] offset
        g0.z = (u32)gaddr;                           // global_addr[31:0]
        g0.w = (u32)(gaddr >> 32) | (2u << 30);      // global_addr[56:32] | type=2
        v8i g1;
        g1[0] = (2 << 16);                           // data_size=2 (4 bytes)
        g1[1] = (int)(((u32)WB & 0xFFFFu) << 16);    // tensor_dim0[15:0] @ [31:16]
        g1[2] = (int)(((u32)WB >> 16) | ((u32)C_ << 16)); // dim0[31:16] | dim1[15:0]
        g1[3] = (int)(((u32)C_ >> 16) | ((u32)TILE_X << 16)); // dim1[31:16] | tile_dim0
        g1[4] = C_;                                  // tile_dim1=64, tile_dim2=0
        g1[5] = HB * WB;                             // tensor_dim0_stride[31:0]
        g1[6] = 0;                                   // stride0[47:32] | stride1[15:0]
        g1[7] = 0;
        v4i gz4  = {0, 0, 0, 0};                     // 2D tensor: groups 2/3 unused
        v8i gz8  = {0, 0, 0, 0, 0, 0, 0, 0};         // trailing group (clang-23 form)
        __builtin_amdgcn_tensor_load_to_lds(g0, g1, gz4, gz4, gz8, 0);
    }

    // ---- Index math (overlaps the DMA) ----
    const float STEP  = 100.0f / 511.0f;             // linspace step for x/y grids
    const float PI_F  = 3.14159265358979f;
    const int   x     = x0 + t;
    const float xf    = -50.0f + (float)x * STEP;
    const float yf    =  50.0f - (float)y * STEP;
    const float phi   = atan2f(yf, xf);
    const float colf  = rintf((phi + PI_F) * ((float)(WR - 1) / (2.0f * PI_F)));
    const int   col   = min(max((int)colf, 0), WR - 1);
    const float rho   = sqrtf(xf * xf + yf * yf) + 1e-6f;
    const int   zb    = zbin[((u64)b * HB + (u64)y) * WB + (u64)x];
    const float zrel  = (float)zb * 0.2f - 3.9f;     // z_bin*dz + (Z_MIN + dz/2)
    const float theta = atan2f(zrel, rho);
    const float TH_MAX   = 0.05235987755983f;        // radians(3)
    const float TH_RANGE = 0.48869219055841f;        // radians(3) - radians(-25)
    const float rowf  = rintf((TH_MAX - theta) * ((float)(HR - 1) / TH_RANGE));
    const int   row   = min(max((int)rowf, 0), HR - 1);
    const int   idx   = row * WR + col;

    if (t < 32) __builtin_amdgcn_s_wait_tensorcnt(0);
    __syncthreads();

    // ---- Scatter-max all 64 channels for this pixel ----
    // Explicit ds_load_b32 batches (8 channels in flight, rows 1KB apart ->
    // lane t hits bank t%64, conflict-free), then 8 device-scope atomics.
    float* obase = out + (u64)b * C_ * NRV + (u64)idx;
    const u32 a0 = lds_base + (u32)t * 4u;           // lane column base (bytes)
#pragma unroll
    for (int cb = 0; cb < C_; cb += 8) {
        float f0, f1, f2, f3, f4, f5, f6, f7;
        u32 a = a0 + (u32)cb * (TILE_X * 4u);
        asm volatile(
            "ds_load_b32 %0, %8 offset:0\n\t"
            "ds_load_b32 %1, %8 offset:1024\n\t"
            "ds_load_b32 %2, %8 offset:2048\n\t"
            "ds_load_b32 %3, %8 offset:3072\n\t"
            "ds_load_b32 %4, %8 offset:4096\n\t"
            "ds_load_b32 %5, %8 offset:5120\n\t"
            "ds_load_b32 %6, %8 offset:6144\n\t"
            "ds_load_b32 %7, %8 offset:7168\n\t"
            "s_wait_dscnt 0x0"
            : "=v"(f0), "=v"(f1), "=v"(f2), "=v"(f3),
              "=v"(f4), "=v"(f5), "=v"(f6), "=v"(f7)
            : "v"(a));
        atomic_max_f32_dev(obase + (u64)(cb + 0) * NRV, f0);
        atomic_max_f32_dev(obase + (u64)(cb + 1) * NRV, f1);
        atomic_max_f32_dev(obase + (u64)(cb + 2) * NRV, f2);
        atomic_max_f32_dev(obase + (u64)(cb + 3) * NRV, f3);
        atomic_max_f32_dev(obase + (u64)(cb + 4) * NRV, f4);
        atomic_max_f32_dev(obase + (u64)(cb + 5) * NRV, f5);
        atomic_max_f32_dev(obase + (u64)(cb + 6) * NRV, f6);
        atomic_max_f32_dev(obase + (u64)(cb + 7) * NRV, f7);
    }
}

__global__ __launch_bounds__(256) void rv_finalize_kernel(float4* __restrict__ out, int n4) {
    int i = blockIdx.x * 256 + threadIdx.x;
    if (i < n4) {
        const float NI = -__builtin_inff();
        float4 v = out[i];
        v.x = (v.x == NI) ? 0.0f : v.x;
        v.y = (v.y == NI) ? 0.0f : v.y;
        v.z = (v.z == NI) ? 0.0f : v.z;
        v.w = (v.w == NI) ? 0.0f : v.w;
        out[i] = v;
    }
}

extern "C" void kernel_launch(void* const* d_in, const int* in_sizes, int n_in,
                              void* d_out, int out_size, void* d_ws, size_t ws_size,
                              hipStream_t stream) {
    const float* bev_feat = (const float*)d_in[0];   // (4,64,512,512) fp32
    const int*   zbin     = (const int*)d_in[1];     // (4,1,512,512) int32
    float*       out      = (float*)d_out;           // (4,64,64,2048) fp32

    const int n  = B_ * C_ * NRV;                    // 33,554,432 floats
    const int n4 = n / 4;                            // 8,388,608 float4

    rv_init_kernel<<<n4 / 256, 256, 0, stream>>>((float4*)out, n4);
    bev2rv_scatter_kernel<<<B_ * HB * (WB / TILE_X), TILE_X, 0, stream>>>(bev_feat, zbin, out);
    rv_finalize_kernel<<<n4 / 256, 256, 0, stream>>>((float4*)out, n4);
}